// MolecularEmbedding_67242007986617
// MI455X (gfx1250) — compile-verified
//
#include <hip/hip_runtime.h>
#include <hip/hip_bf16.h>

// out[b,s,:] = relu(W1[x[b,s]] + b1) @ W2 + b2
// Key insight: result depends only on x's value (0..127) -> precompute a
// 128x64 table with full-f32 WMMA, then do a bandwidth-bound gather.

typedef __attribute__((ext_vector_type(2))) float v2f;
typedef __attribute__((ext_vector_type(8))) float v8f;

// ---------------------------------------------------------------------------
// Kernel A: T[e, o] = relu(W1[e, :] + b1) @ W2[:, o] + b2[o]
// One wave (32 threads) per 16x16 output tile; 8 M-tiles x 4 N-tiles = 32 blocks.
// Uses V_WMMA_F32_16X16X4_F32 (full f32 precision, matches reference numerics).
//
// ISA layouts (CDNA5 ISA §7.12.2), wave32:
//   A 16x4 f32 (2 VGPRs): lane L holds M = L&15; v0 -> K = (L<16 ? 0 : 2),
//                         v1 -> K = (L<16 ? 1 : 3)
//   B 4x16 f32 (2 VGPRs): lane L holds N = L&15; v0 -> K row (L<16 ? 0 : 2),
//                         v1 -> K row (L<16 ? 1 : 3)
//   C/D 16x16 f32 (8 VGPRs): VGPR r, lane L -> (M = r + (L<16?0:8), N = L&15)
// ---------------------------------------------------------------------------
__global__ __launch_bounds__(32) void me_build_table(
    const float* __restrict__ W1,   // [128,128]
    const float* __restrict__ b1,   // [128]
    const float* __restrict__ W2,   // [128,64]
    const float* __restrict__ b2,   // [64]
    float* __restrict__ T)          // [128,64] (workspace)
{
    const int lane = threadIdx.x;        // 0..31, full wave, EXEC all-1s
    const int tile = blockIdx.x;         // 0..31
    const int mt   = tile >> 2;          // 0..7   (rows of T, element ids)
    const int nt   = tile & 3;           // 0..3   (cols of T, outputs)

    const int m   = mt * 16 + (lane & 15);   // A-fragment row (element id)
    const int n   = nt * 16 + (lane & 15);   // B/C/D-fragment column
    const int khi = (lane >> 4) << 1;        // 0 for lanes 0-15, 2 for 16-31

    // Seed accumulator with bias b2 (same value in all 8 row-VGPRs per lane).
    v8f c;
    const float beta = b2[n];
#pragma unroll
    for (int r = 0; r < 8; ++r) c[r] = beta;

    // K = 128 in steps of 4.
#pragma unroll 4
    for (int kk = 0; kk < 32; ++kk) {
        const int k0 = kk * 4 + khi;
        const int k1 = k0 + 1;

        v2f a, b;
        a.x = fmaxf(W1[m * 128 + k0] + b1[k0], 0.0f);
        a.y = fmaxf(W1[m * 128 + k1] + b1[k1], 0.0f);
        b.x = W2[k0 * 64 + n];
        b.y = W2[k1 * 64 + n];

        // 8 args: (neg_a, A, neg_b, B, c_mod, C, reuse_a, reuse_b)
        c = __builtin_amdgcn_wmma_f32_16x16x4_f32(
            false, a, false, b, (short)0, c, false, false);
    }

    // Scatter D back per ISA C/D layout.
    const int mbase = mt * 16 + ((lane >> 4) << 3);  // +8 for upper half-wave
#pragma unroll
    for (int r = 0; r < 8; ++r) {
        T[(mbase + r) * 64 + n] = c[r];
    }
}

// ---------------------------------------------------------------------------
// Kernel B: out[tok, :] = T[x[tok], :]
// Pure bandwidth: 2 MB index reads + 134 MB fp32 writes -> ~5.9 us @ 23.3 TB/s.
// Table staged in LDS (32 KB of the WGP's 320 KB); each 16-lane group writes
// one 64-float row as 16 coalesced float4 (global_store_b128) stores.
// ---------------------------------------------------------------------------
__global__ __launch_bounds__(256) void me_gather(
    const int* __restrict__ x,      // [ntok] element ids
    const float* __restrict__ T,    // [128,64] table
    float* __restrict__ out,        // [ntok,64]
    int ntok)
{
    __shared__ float4 Tlds[128 * 16];       // 128 rows x 16 float4 = 32 KB

    const float4* __restrict__ Tv = (const float4*)T;
    for (int i = threadIdx.x; i < 128 * 16; i += 256) Tlds[i] = Tv[i];
    __syncthreads();

    const int group  = threadIdx.x >> 4;    // 0..15 : token within block-iter
    const int lane16 = threadIdx.x & 15;    // 0..15 : float4 within row
    float4* __restrict__ outv = (float4*)out;

    for (int base = blockIdx.x * 16; base < ntok; base += gridDim.x * 16) {
        const int tok = base + group;
        if (tok < ntok) {
            const int id = x[tok];          // broadcast load within group
            outv[tok * 16 + lane16] = Tlds[id * 16 + lane16];
        }
    }
}

extern "C" void kernel_launch(void* const* d_in, const int* in_sizes, int n_in,
                              void* d_out, int out_size, void* d_ws, size_t ws_size,
                              hipStream_t stream) {
    const int*   x  = (const int*)  d_in[0];   // [1024,512] int ids
    const float* W1 = (const float*)d_in[1];   // [128,128]
    const float* b1 = (const float*)d_in[2];   // [128]
    const float* W2 = (const float*)d_in[3];   // [128,64]
    const float* b2 = (const float*)d_in[4];   // [64]
    float* out = (float*)d_out;                // [1024,512,64]
    float* T   = (float*)d_ws;                 // 32 KB table scratch

    const int ntok = in_sizes[0];              // 524288

    me_build_table<<<32, 32, 0, stream>>>(W1, b1, W2, b2, T);

    int ngroups = (ntok + 15) / 16;
    int blocks  = ngroups < 2048 ? ngroups : 2048;
    me_gather<<<blocks, 256, 0, stream>>>(x, T, out, ntok);
}